// GAT_57664230916770
// MI455X (gfx1250) — compile-verified
//
#include <hip/hip_runtime.h>
#include <hip/hip_bf16.h>

// ---------------- problem constants ----------------
#define NN      10000
#define EE      160000
#define F_NODE  128
#define F_BOND  16
#define HH      4
#define CC      64
#define HC      256
#define DEPTH   5
#define NEG     0.2f
#define NPAD    10016        // 313 * 32, padded row count for GEMM tiling

typedef __attribute__((ext_vector_type(2))) float v2f;
typedef __attribute__((ext_vector_type(8))) float v8f;

// async-copy helpers (CDNA5: memory -> LDS direct, tracked by ASYNCcnt)
__device__ __forceinline__ void async_b64_to_lds(unsigned ldsAddr, const void* gptr) {
    unsigned long long ga = (unsigned long long)(size_t)gptr;
    asm volatile("global_load_async_to_lds_b64 %0, %1, off"
                 :: "v"(ldsAddr), "v"(ga) : "memory");
}
__device__ __forceinline__ void wait_asynccnt0() {
#if __has_builtin(__builtin_amdgcn_s_wait_asynccnt)
    __builtin_amdgcn_s_wait_asynccnt(0);
#else
    asm volatile("s_wait_asynccnt 0x0" ::: "memory");
#endif
}

// ---------------- fill helpers ----------------
__global__ void gat_fillf(float* p, float v, int n) {
    int i = blockIdx.x * blockDim.x + threadIdx.x;
    if (i < n) p[i] = v;
}
__global__ void gat_fillu(unsigned* p, unsigned v, int n) {
    int i = blockIdx.x * blockDim.x + threadIdx.x;
    if (i < n) p[i] = v;
}

// ---------------- self-loop edge-attr (mean of incoming) ----------------
__global__ void gat_deg_attr(const int* __restrict__ dstI,
                             const float* __restrict__ edge_attr,
                             float* __restrict__ deg, float* __restrict__ loop_sum) {
    int e = blockIdx.x * blockDim.x + threadIdx.x;
    if (e >= EE) return;
    int d = dstI[e];
    atomicAdd(&deg[d], 1.0f);
    #pragma unroll
    for (int b = 0; b < F_BOND; ++b)
        atomicAdd(&loop_sum[d * F_BOND + b], edge_attr[e * F_BOND + b]);
}
__global__ void gat_loop_norm(float* __restrict__ loop_attr, const float* __restrict__ deg) {
    int i = blockIdx.x * blockDim.x + threadIdx.x;
    if (i >= NN * F_BOND) return;
    float dg = deg[i >> 4];
    if (dg < 1.0f) dg = 1.0f;
    loop_attr[i] = loop_attr[i] / dg;
}

// ---------------- fp32 WMMA GEMM: D[N,256] = A[N,K] @ B[K,256] ----------------
// Block = 256 threads = 8 waves arranged 2x4; block tile 32 rows x 64 cols.
// A tile async-copied memory->LDS (2KB/chunk = 256 thr x b64), stride-17 pad
// gives bank-conflict-free fragment reads. B prefetched one chunk ahead.
__global__ __launch_bounds__(256)
void gat_gemm_xw(const float* __restrict__ A, int K,
                 const float* __restrict__ B,
                 float* __restrict__ D, int nrows) {
    __shared__ float sA[32][17];
    const int tid  = threadIdx.x;
    const int lane = tid & 31;
    const int wave = tid >> 5;
    const int wm = wave >> 2;              // 0..1
    const int wn = wave & 3;               // 0..3
    const int rowBlk = blockIdx.x * 32;
    const int col0   = blockIdx.y * 64 + wn * 16;
    const int mloc   = wm * 16 + (lane & 15);
    const int nn     = col0 + (lane & 15);
    const int khalf  = (lane >> 4) * 2;    // 0 or 2

    // per-thread async-copy src/dst: thread covers 8B of the 32x16 A chunk
    const int i0 = tid * 2;                // 0..510, pairs share a row
    const int r  = i0 >> 4, c = i0 & 15;
    int gr = rowBlk + r; if (gr >= nrows) gr = nrows - 1;   // branchless clamp
    const float*   gsrc   = A + (size_t)gr * K + c;
    const unsigned ldsDst = (unsigned)(size_t)&sA[r][c];

    v8f acc = {};
    for (int kc = 0; kc < K; kc += 16) {
        async_b64_to_lds(ldsDst, gsrc + kc);
        if (kc + 16 < K)                   // prefetch next B chunk (global_prefetch_b8)
            __builtin_prefetch(B + (kc + 16) * HC + nn, 0, 1);
        wait_asynccnt0();
        __syncthreads();
        #pragma unroll
        for (int ks = 0; ks < 16; ks += 4) {
            const int kl = ks + khalf;
            v2f a, b;
            a.x = sA[mloc][kl];
            a.y = sA[mloc][kl + 1];
            const int kg = kc + kl;
            b.x = B[kg * HC + nn];
            b.y = B[(kg + 1) * HC + nn];
            acc = __builtin_amdgcn_wmma_f32_16x16x4_f32(
                    false, a, false, b, (short)0, acc, false, false);
        }
        __syncthreads();
    }
    const int mbase = rowBlk + wm * 16 + (lane >> 4) * 8;
    #pragma unroll
    for (int v = 0; v < 8; ++v) {
        int mr = mbase + v;
        if (mr < nrows) D[(size_t)mr * HC + nn] = acc[v];
    }
}

// ---------------- per-node attention scores a_s, a_d ----------------
__global__ __launch_bounds__(256)
void gat_node_scores(const float* __restrict__ xw,
                     const float* __restrict__ att_src,   // [4][64]
                     const float* __restrict__ att_dst,   // [4][64]
                     float* __restrict__ a_s, float* __restrict__ a_d) {
    int wid  = (blockIdx.x * blockDim.x + threadIdx.x) >> 5;
    int lane = threadIdx.x & 31;
    int node = wid >> 2, h = wid & 3;
    if (node >= NN) return;
    const float* xp = xw + (size_t)node * HC + h * CC;
    const float* as = att_src + h * CC;
    const float* ad = att_dst + h * CC;
    float s1 = xp[lane] * as[lane] + xp[lane + 32] * as[lane + 32];
    float s2 = xp[lane] * ad[lane] + xp[lane + 32] * ad[lane + 32];
    #pragma unroll
    for (int m = 16; m; m >>= 1) {
        s1 += __shfl_xor(s1, m, 32);
        s2 += __shfl_xor(s2, m, 32);
    }
    if (lane == 0) { a_s[node * HH + h] = s1; a_d[node * HH + h] = s2; }
}

// ---------------- fold Wedge . att_edge -> Q[16][4] ----------------
__global__ void gat_foldq(const float* __restrict__ Wedge,     // [16][256]
                          const float* __restrict__ att_edge,  // [4][64]
                          float* __restrict__ Q) {
    int t = threadIdx.x;                  // 64 threads
    int b = t >> 2, h = t & 3;
    float s = 0.0f;
    #pragma unroll 8
    for (int c = 0; c < CC; ++c) s += Wedge[b * HC + h * CC + c] * att_edge[h * CC + c];
    Q[b * HH + h] = s;
}

// ---------------- per-edge alpha + leaky-relu + segment max ----------------
__global__ __launch_bounds__(256)
void gat_alpha_max(const float* __restrict__ a_s, const float* __restrict__ a_d,
                   const float* __restrict__ Q,
                   const float* __restrict__ edge_attr, const float* __restrict__ loop_attr,
                   const int* __restrict__ srcI, const int* __restrict__ dstI,
                   float* __restrict__ alpha, unsigned* __restrict__ mmax) {
    int e = blockIdx.x * blockDim.x + threadIdx.x;
    if (e >= EE + NN) return;
    int s, d; const float* ea;
    if (e < EE) { s = srcI[e]; d = dstI[e]; ea = edge_attr + (size_t)e * F_BOND; }
    else        { s = d = e - EE;           ea = loop_attr + (size_t)s * F_BOND; }
    float eav[F_BOND];
    #pragma unroll
    for (int b = 0; b < F_BOND; ++b) eav[b] = ea[b];
    #pragma unroll
    for (int h = 0; h < HH; ++h) {
        float ae = 0.0f;
        #pragma unroll
        for (int b = 0; b < F_BOND; ++b) ae += eav[b] * Q[b * HH + h];
        float al = a_s[s * HH + h] + a_d[d * HH + h] + ae;
        al = (al > 0.0f) ? al : al * NEG;
        alpha[(size_t)e * HH + h] = al;
        unsigned ub  = __float_as_uint(al);
        unsigned key = (ub & 0x80000000u) ? ~ub : (ub | 0x80000000u);  // order-preserving
        atomicMax(&mmax[d * HH + h], key);
    }
}

// ---------------- exp(alpha - max) + segment sum ----------------
__global__ __launch_bounds__(256)
void gat_expsum(float* __restrict__ alpha, const unsigned* __restrict__ mmax,
                const int* __restrict__ dstI, float* __restrict__ denom) {
    int e = blockIdx.x * blockDim.x + threadIdx.x;
    if (e >= EE + NN) return;
    int d = (e < EE) ? dstI[e] : (e - EE);
    #pragma unroll
    for (int h = 0; h < HH; ++h) {
        unsigned key = mmax[d * HH + h];
        unsigned ub  = (key & 0x80000000u) ? (key & 0x7FFFFFFFu) : ~key;
        float m  = __uint_as_float(ub);
        float ex = expf(alpha[(size_t)e * HH + h] - m);
        alpha[(size_t)e * HH + h] = ex;
        atomicAdd(&denom[d * HH + h], ex);
    }
}

// ---------------- message scatter: one wave per edge ----------------
__global__ __launch_bounds__(256)
void gat_scatter(const float* __restrict__ xw, const float* __restrict__ alpha,
                 const float* __restrict__ denom,
                 const int* __restrict__ srcI, const int* __restrict__ dstI,
                 float* __restrict__ agg) {
    int wid = (blockIdx.x * blockDim.x + threadIdx.x) >> 5;
    if (wid >= EE + NN) return;
    int lane = threadIdx.x & 31;
    int s, d;
    if (wid < EE) { s = srcI[wid]; d = dstI[wid]; } else { s = d = wid - EE; }
    float co[HH];
    #pragma unroll
    for (int h = 0; h < HH; ++h)
        co[h] = alpha[(size_t)wid * HH + h] / denom[d * HH + h];
    const float* xp = xw + (size_t)s * HC;
    float* ap = agg + (size_t)d * HC;
    #pragma unroll
    for (int j = 0; j < 8; ++j) {
        int c = lane + j * 32;            // head index = j>>1, compile-time after unroll
        atomicAdd(&ap[c], xp[c] * co[j >> 1]);
    }
}

// ---------------- bias + relu ----------------
__global__ void gat_bias_relu(const float* __restrict__ agg, const float* __restrict__ bias,
                              float* __restrict__ h) {
    int i = blockIdx.x * blockDim.x + threadIdx.x;
    if (i >= NN * HC) return;
    float v = agg[i] + bias[i & (HC - 1)];
    h[i] = (v > 0.0f) ? v : 0.0f;
}

// ---------------- final FC: relu([x | h] @ fc_w + fc_b), K = 384 ----------------
__global__ __launch_bounds__(256)
void gat_gemm_final(const float* __restrict__ x, const float* __restrict__ h,
                    const float* __restrict__ fcw,    // [384][64]
                    const float* __restrict__ fcb,    // [64]
                    float* __restrict__ out) {
    __shared__ float sA[32][17];
    const int tid  = threadIdx.x;
    const int lane = tid & 31;
    const int wave = tid >> 5;
    const int wm = wave >> 2, wn = wave & 3;
    const int rowBlk = blockIdx.x * 32;
    const int col0   = wn * 16;            // 64 output cols covered by one block
    const int mloc   = wm * 16 + (lane & 15);
    const int nn     = col0 + (lane & 15);
    const int khalf  = (lane >> 4) * 2;
    const int K = F_NODE + HC;             // 384

    const int i0 = tid * 2;
    const int r  = i0 >> 4, c = i0 & 15;
    int gr = rowBlk + r; if (gr >= NN) gr = NN - 1;
    const float*   gx     = x + (size_t)gr * F_NODE + c;
    const float*   gh     = h + (size_t)gr * HC + c;
    const unsigned ldsDst = (unsigned)(size_t)&sA[r][c];

    v8f acc = {};
    for (int kc = 0; kc < K; kc += 16) {
        const float* gsrc = (kc < F_NODE) ? (gx + kc) : (gh + (kc - F_NODE));
        async_b64_to_lds(ldsDst, gsrc);
        if (kc + 16 < K)
            __builtin_prefetch(fcw + (kc + 16) * CC + nn, 0, 1);
        wait_asynccnt0();
        __syncthreads();
        #pragma unroll
        for (int ks = 0; ks < 16; ks += 4) {
            const int kl = ks + khalf;
            v2f a, b;
            a.x = sA[mloc][kl];
            a.y = sA[mloc][kl + 1];
            const int kg = kc + kl;
            b.x = fcw[kg * CC + nn];
            b.y = fcw[(kg + 1) * CC + nn];
            acc = __builtin_amdgcn_wmma_f32_16x16x4_f32(
                    false, a, false, b, (short)0, acc, false, false);
        }
        __syncthreads();
    }
    const int mbase = rowBlk + wm * 16 + (lane >> 4) * 8;
    const float bb = fcb[nn];
    #pragma unroll
    for (int v = 0; v < 8; ++v) {
        int mr = mbase + v;
        if (mr < NN) {
            float rr = acc[v] + bb;
            out[(size_t)mr * CC + nn] = (rr > 0.0f) ? rr : 0.0f;
        }
    }
}

// ---------------- host orchestration ----------------
extern "C" void kernel_launch(void* const* d_in, const int* in_sizes, int n_in,
                              void* d_out, int out_size, void* d_ws, size_t ws_size,
                              hipStream_t stream) {
    (void)in_sizes; (void)n_in; (void)out_size; (void)ws_size;
    const float* x         = (const float*)d_in[0];
    const int*   edge_idx  = (const int*)  d_in[1];
    const float* edge_attr = (const float*)d_in[2];
    const float* W0        = (const float*)d_in[3];
    const float* Ws        = (const float*)d_in[4];
    const float* att_src   = (const float*)d_in[5];
    const float* att_dst   = (const float*)d_in[6];
    const float* Wedge     = (const float*)d_in[7];
    const float* att_edge  = (const float*)d_in[8];
    const float* biases    = (const float*)d_in[9];
    const float* fc_w      = (const float*)d_in[10];
    const float* fc_b      = (const float*)d_in[11];
    float* out = (float*)d_out;

    const int* srcI = edge_idx;
    const int* dstI = edge_idx + EE;

    // workspace carve (floats)
    float* f = (float*)d_ws;
    float*    xw        = f;                    f += (size_t)NPAD * HC;
    float*    agg       = f;                    f += (size_t)NPAD * HC;
    float*    h_cur     = f;                    f += (size_t)NPAD * HC;
    float*    a_s       = f;                    f += NN * HH;
    float*    a_d       = f;                    f += NN * HH;
    float*    alpha     = f;                    f += (size_t)(EE + NN) * HH;
    float*    denom     = f;                    f += NN * HH;
    float*    Q         = f;                    f += F_BOND * HH;
    float*    deg       = f;                    f += NN;
    float*    loop_attr = f;                    f += NN * F_BOND;
    unsigned* mmax      = (unsigned*)f;

    const int T = 256;
    const int EN = EE + NN;

    // self-loop mean edge attributes (deterministic, recomputed every call)
    gat_fillf<<<(NN + T - 1) / T, T, 0, stream>>>(deg, 0.0f, NN);
    gat_fillf<<<(NN * F_BOND + T - 1) / T, T, 0, stream>>>(loop_attr, 0.0f, NN * F_BOND);
    gat_deg_attr<<<(EE + T - 1) / T, T, 0, stream>>>(dstI, edge_attr, deg, loop_attr);
    gat_loop_norm<<<(NN * F_BOND + T - 1) / T, T, 0, stream>>>(loop_attr, deg);

    const float* hin = x;
    int K = F_NODE;
    for (int i = 0; i < DEPTH; ++i) {
        const float* W = (i == 0) ? W0 : (Ws + (size_t)(i - 1) * HC * HC);
        gat_gemm_xw<<<dim3(NPAD / 32, HC / 64), T, 0, stream>>>(hin, K, W, xw, NN);
        gat_node_scores<<<(NN * HH + 7) / 8, T, 0, stream>>>(
            xw, att_src + i * HH * CC, att_dst + i * HH * CC, a_s, a_d);
        gat_foldq<<<1, F_BOND * HH, 0, stream>>>(
            Wedge + (size_t)i * F_BOND * HC, att_edge + i * HH * CC, Q);
        gat_fillu<<<(NN * HH + T - 1) / T, T, 0, stream>>>(mmax, 0u, NN * HH);
        gat_fillf<<<(NN * HH + T - 1) / T, T, 0, stream>>>(denom, 0.0f, NN * HH);
        gat_fillf<<<(NN * HC + T - 1) / T, T, 0, stream>>>(agg, 0.0f, NN * HC);
        gat_alpha_max<<<(EN + T - 1) / T, T, 0, stream>>>(
            a_s, a_d, Q, edge_attr, loop_attr, srcI, dstI, alpha, mmax);
        gat_expsum<<<(EN + T - 1) / T, T, 0, stream>>>(alpha, mmax, dstI, denom);
        gat_scatter<<<(EN + 7) / 8, T, 0, stream>>>(xw, alpha, denom, srcI, dstI, agg);
        gat_bias_relu<<<(NN * HC + T - 1) / T, T, 0, stream>>>(agg, biases + i * HC, h_cur);
        hin = h_cur;
        K = HC;
    }
    gat_gemm_final<<<dim3(NPAD / 32, 1), T, 0, stream>>>(x, h_cur, fc_w, fc_b, out);
}